// GCN_30296699306159
// MI455X (gfx1250) — compile-verified
//
#include <hip/hip_runtime.h>
#include <hip/hip_bf16.h>

#define N_NODES 30000
#define E_RAW   300000
#define E_TOT   330000
#define HEADS   8
#define HID     64
#define F1      512          // HEADS*HID
#define IN_DIM  8
#define OUT_DIM 8
#define NEG_SLOPE 0.2f
#define M_TILES 1875         // 30000/16

typedef __attribute__((ext_vector_type(2))) float v2f;
typedef __attribute__((ext_vector_type(8))) float v8f;

// ---------- helpers ----------
__device__ __forceinline__ unsigned fmap(float f) {
    unsigned u = __float_as_uint(f);
    return (u & 0x80000000u) ? ~u : (u | 0x80000000u);
}
__device__ __forceinline__ float funmap(unsigned u) {
    return __uint_as_float((u & 0x80000000u) ? (u ^ 0x80000000u) : ~u);
}
__device__ __forceinline__ void get_edge(const int* __restrict__ ei, int e, int& s, int& d) {
    if (e < E_RAW) { s = ei[e]; d = ei[E_RAW + e]; }
    else           { s = e - E_RAW; d = e - E_RAW; }
}
__device__ __forceinline__ float lrelu(float v) {
    return v > 0.f ? v : v * NEG_SLOPE;
}

// ---------- init kernels ----------
__global__ void k_init1(unsigned* __restrict__ m1, float* __restrict__ s1) {
    int i = blockIdx.x * blockDim.x + threadIdx.x;
    if (i < N_NODES * HEADS) { m1[i] = 0u; s1[i] = 0.f; }
}
__global__ void k_zero4(float4* __restrict__ p, int n4) {
    int i = blockIdx.x * blockDim.x + threadIdx.x;
    if (i < n4) p[i] = make_float4(0.f, 0.f, 0.f, 0.f);
}
__global__ void k_init2(float* __restrict__ out, const float* __restrict__ b2,
                        unsigned* __restrict__ m2, float* __restrict__ s2) {
    int i = blockIdx.x * blockDim.x + threadIdx.x;
    if (i < N_NODES * OUT_DIM) out[i] = b2[i & 7];
    if (i < N_NODES) { m2[i] = 0u; s2[i] = 0.f; }
}

// ---------- layer-1 GEMM: xl = x@Wl, xr = x@Wr  via V_WMMA_F32_16X16X4_F32 ----------
// grid (1875, 4, 2), block 256 (8 waves; wave w -> N-tile blockIdx.y*8+w)
__global__ void k_gemm1(const float* __restrict__ x,   // [30000,8]
                        const float* __restrict__ Wl,  // [8,512]
                        const float* __restrict__ Wr,  // [8,512]
                        float* __restrict__ xl, float* __restrict__ xr) {
    const int wave  = threadIdx.x >> 5;
    const int lane  = threadIdx.x & 31;
    const int row   = lane & 15;
    const int khalf = lane >> 4;
    const int mbase = blockIdx.x * 16;
    const int nbase = (blockIdx.y * 8 + wave) * 16;
    const float* __restrict__ W = blockIdx.z ? Wr : Wl;
    float* __restrict__ O       = blockIdx.z ? xr : xl;

    v8f c = {};
#pragma unroll
    for (int kk = 0; kk < 2; ++kk) {
        const int k0 = kk * 4 + khalf * 2;
        v2f a, b;
        a.x = x[(mbase + row) * IN_DIM + k0];
        a.y = x[(mbase + row) * IN_DIM + k0 + 1];
        b.x = W[(k0    ) * F1 + nbase + row];
        b.y = W[(k0 + 1) * F1 + nbase + row];
        c = __builtin_amdgcn_wmma_f32_16x16x4_f32(false, a, false, b,
                                                  (short)0, c, false, false);
    }
#pragma unroll
    for (int r = 0; r < 8; ++r)
        O[(size_t)(mbase + r + 8 * khalf) * F1 + nbase + row] = c[r];
}

// ---------- layer-1 edge logits + segment max (wave per edge) ----------
__global__ void k_edge1_logits(const int* __restrict__ ei,
                               const float* __restrict__ xl, const float* __restrict__ xr,
                               const float* __restrict__ a1,   // [8,64] flat
                               float* __restrict__ e1, unsigned* __restrict__ m1) {
    const int wid  = (int)((blockIdx.x * blockDim.x + threadIdx.x) >> 5);
    const int lane = threadIdx.x & 31;
    if (wid >= E_TOT) return;
    int s, d; get_edge(ei, wid, s, d);
    const float4* __restrict__ pl = (const float4*)(xl + (size_t)s * F1) + lane * 4;
    const float4* __restrict__ pr = (const float4*)(xr + (size_t)d * F1) + lane * 4;
    const float4* __restrict__ pa = (const float4*)a1 + lane * 4;
    float acc = 0.f;
#pragma unroll
    for (int i = 0; i < 4; ++i) {
        float4 l = pl[i], r = pr[i], a = pa[i];
        acc = fmaf(lrelu(l.x + r.x), a.x, acc);
        acc = fmaf(lrelu(l.y + r.y), a.y, acc);
        acc = fmaf(lrelu(l.z + r.z), a.z, acc);
        acc = fmaf(lrelu(l.w + r.w), a.w, acc);
    }
    acc += __shfl_xor(acc, 1, 32);
    acc += __shfl_xor(acc, 2, 32);
    if ((lane & 3) == 0) {
        const int h = lane >> 2;
        e1[(size_t)wid * HEADS + h] = acc;
        atomicMax(&m1[d * HEADS + h], fmap(acc));
    }
}

// ---------- layer-1 exp + denom (thread per edge*head) ----------
__global__ void k_edge1_exp(const int* __restrict__ ei, float* __restrict__ e1,
                            const unsigned* __restrict__ m1, float* __restrict__ s1) {
    int idx = blockIdx.x * blockDim.x + threadIdx.x;
    if (idx >= E_TOT * HEADS) return;
    const int e = idx >> 3, h = idx & 7;
    int s, d; get_edge(ei, e, s, d);
    const float p = __expf(e1[idx] - funmap(m1[d * HEADS + h]));
    e1[idx] = p;
    atomicAdd(&s1[d * HEADS + h], p);
}

// ---------- layer-1 aggregation (wave per edge, atomic f32 adds) ----------
__global__ void k_edge1_agg(const int* __restrict__ ei,
                            const float* __restrict__ xl,
                            const float* __restrict__ e1, const float* __restrict__ s1,
                            float* __restrict__ h1) {
    const int wid  = (int)((blockIdx.x * blockDim.x + threadIdx.x) >> 5);
    const int lane = threadIdx.x & 31;
    if (wid >= E_TOT) return;
    int s, d; get_edge(ei, wid, s, d);
    const int h = lane >> 2;
    const float alpha = e1[(size_t)wid * HEADS + h] / s1[d * HEADS + h];
    const float4* __restrict__ pl = (const float4*)(xl + (size_t)s * F1) + lane * 4;
    float* __restrict__ po = h1 + (size_t)d * F1 + lane * 16;
#pragma unroll
    for (int i = 0; i < 4; ++i) {
        float4 l = pl[i];
        atomicAdd(po + i * 4 + 0, alpha * l.x);
        atomicAdd(po + i * 4 + 1, alpha * l.y);
        atomicAdd(po + i * 4 + 2, alpha * l.z);
        atomicAdd(po + i * 4 + 3, alpha * l.w);
    }
}

// ---------- bias + relu (in place) ----------
__global__ void k_bias_relu(float* __restrict__ h1, const float* __restrict__ b1) {
    int i = blockIdx.x * blockDim.x + threadIdx.x;
    if (i >= N_NODES * F1) return;
    const float v = h1[i] + b1[i & (F1 - 1)];
    h1[i] = v > 0.f ? v : 0.f;
}

// ---------- layer-2 GEMM: h2l = h@W2l, h2r = h@W2r  (WMMA f32 16x16x4, K=512) ----------
// grid(469), block 256; wave w -> mtile = blockIdx.x*4 + (w>>1), matrix = w&1
__global__ void k_gemm2(const float* __restrict__ h,    // [30000,512]
                        const float* __restrict__ W2l,  // [512,8]
                        const float* __restrict__ W2r,
                        float* __restrict__ h2l, float* __restrict__ h2r) {
    const int wave  = threadIdx.x >> 5;
    const int lane  = threadIdx.x & 31;
    const int row   = lane & 15;
    const int khalf = lane >> 4;
    const int mtile = blockIdx.x * 4 + (wave >> 1);
    if (mtile >= M_TILES) return;
    const int mbase = mtile * 16;
    const float* __restrict__ W = (wave & 1) ? W2r : W2l;
    float* __restrict__ O       = (wave & 1) ? h2r : h2l;

    v8f c = {};
    for (int kk = 0; kk < F1 / 4; ++kk) {
        const int k0 = kk * 4 + khalf * 2;
        v2f a, b;
        a.x = h[(size_t)(mbase + row) * F1 + k0];
        a.y = h[(size_t)(mbase + row) * F1 + k0 + 1];
        b.x = (row < OUT_DIM) ? W[(k0    ) * OUT_DIM + row] : 0.f;
        b.y = (row < OUT_DIM) ? W[(k0 + 1) * OUT_DIM + row] : 0.f;
        c = __builtin_amdgcn_wmma_f32_16x16x4_f32(false, a, false, b,
                                                  (short)0, c, false, false);
    }
    if (row < OUT_DIM) {
#pragma unroll
        for (int r = 0; r < 8; ++r)
            O[(size_t)(mbase + r + 8 * khalf) * OUT_DIM + row] = c[r];
    }
}

// ---------- layer-2 edge logits (thread per edge) ----------
__global__ void k_edge2_logits(const int* __restrict__ ei,
                               const float* __restrict__ h2l, const float* __restrict__ h2r,
                               const float* __restrict__ a2,
                               float* __restrict__ e2, unsigned* __restrict__ m2) {
    int e = blockIdx.x * blockDim.x + threadIdx.x;
    if (e >= E_TOT) return;
    int s, d; get_edge(ei, e, s, d);
    float acc = 0.f;
#pragma unroll
    for (int c = 0; c < OUT_DIM; ++c)
        acc = fmaf(lrelu(h2l[s * OUT_DIM + c] + h2r[d * OUT_DIM + c]), a2[c], acc);
    e2[e] = acc;
    atomicMax(&m2[d], fmap(acc));
}

__global__ void k_edge2_exp(const int* __restrict__ ei, float* __restrict__ e2,
                            const unsigned* __restrict__ m2, float* __restrict__ s2) {
    int e = blockIdx.x * blockDim.x + threadIdx.x;
    if (e >= E_TOT) return;
    int s, d; get_edge(ei, e, s, d);
    const float p = __expf(e2[e] - funmap(m2[d]));
    e2[e] = p;
    atomicAdd(&s2[d], p);
}

__global__ void k_edge2_agg(const int* __restrict__ ei,
                            const float* __restrict__ h2l,
                            const float* __restrict__ e2, const float* __restrict__ s2,
                            float* __restrict__ out) {
    int idx = blockIdx.x * blockDim.x + threadIdx.x;
    if (idx >= E_TOT * OUT_DIM) return;
    const int e = idx >> 3, c = idx & 7;
    int s, d; get_edge(ei, e, s, d);
    const float alpha = e2[e] / s2[d];
    atomicAdd(&out[d * OUT_DIM + c], alpha * h2l[s * OUT_DIM + c]);
}

// ---------- host launch ----------
extern "C" void kernel_launch(void* const* d_in, const int* in_sizes, int n_in,
                              void* d_out, int out_size, void* d_ws, size_t ws_size,
                              hipStream_t stream) {
    const float* x   = (const float*)d_in[0];
    const int*   ei  = (const int*)  d_in[1];
    const float* W1l = (const float*)d_in[2];
    const float* W1r = (const float*)d_in[3];
    const float* a1  = (const float*)d_in[4];
    const float* b1  = (const float*)d_in[5];
    const float* W2l = (const float*)d_in[6];
    const float* W2r = (const float*)d_in[7];
    const float* a2  = (const float*)d_in[8];
    const float* b2  = (const float*)d_in[9];
    float* out = (float*)d_out;

    char* ws = (char*)d_ws;
    size_t off = 0;
    auto carve = [&](size_t bytes) -> char* {
        char* p = ws + off;
        off += (bytes + 255) & ~(size_t)255;
        return p;
    };
    float*    xl1 = (float*)   carve((size_t)N_NODES * F1 * 4);
    float*    xr1 = (float*)   carve((size_t)N_NODES * F1 * 4);  // reused as h1
    float*    e1  = (float*)   carve((size_t)E_TOT * HEADS * 4);
    unsigned* m1  = (unsigned*)carve((size_t)N_NODES * HEADS * 4);
    float*    s1  = (float*)   carve((size_t)N_NODES * HEADS * 4);
    float*    h2l = (float*)   carve((size_t)N_NODES * OUT_DIM * 4);
    float*    h2r = (float*)   carve((size_t)N_NODES * OUT_DIM * 4);
    float*    e2  = (float*)   carve((size_t)E_TOT * 4);
    unsigned* m2  = (unsigned*)carve((size_t)N_NODES * 4);
    float*    s2  = (float*)   carve((size_t)N_NODES * 4);
    float*    h1  = xr1;  // alias: xr1 dead after k_edge1_logits
    (void)ws_size; (void)n_in; (void)in_sizes; (void)out_size;

    const int B = 256;

    // ---- layer 1 ----
    k_init1<<<(N_NODES * HEADS + B - 1) / B, B, 0, stream>>>(m1, s1);
    k_gemm1<<<dim3(M_TILES, 4, 2), B, 0, stream>>>(x, W1l, W1r, xl1, xr1);
    {   // wave per edge
        int waves = E_TOT, blocks = (waves * 32 + B - 1) / B;
        k_edge1_logits<<<blocks, B, 0, stream>>>(ei, xl1, xr1, a1, e1, m1);
    }
    k_edge1_exp<<<(E_TOT * HEADS + B - 1) / B, B, 0, stream>>>(ei, e1, m1, s1);
    k_zero4<<<((N_NODES * F1 / 4) + B - 1) / B, B, 0, stream>>>((float4*)h1, N_NODES * F1 / 4);
    {
        int waves = E_TOT, blocks = (waves * 32 + B - 1) / B;
        k_edge1_agg<<<blocks, B, 0, stream>>>(ei, xl1, e1, s1, h1);
    }
    k_bias_relu<<<(N_NODES * F1 + B - 1) / B, B, 0, stream>>>(h1, b1);

    // ---- layer 2 ----
    k_gemm2<<<(M_TILES + 3) / 4, B, 0, stream>>>(h1, W2l, W2r, h2l, h2r);
    k_init2<<<(N_NODES * OUT_DIM + B - 1) / B, B, 0, stream>>>(out, b2, m2, s2);
    k_edge2_logits<<<(E_TOT + B - 1) / B, B, 0, stream>>>(ei, h2l, h2r, a2, e2, m2);
    k_edge2_exp<<<(E_TOT + B - 1) / B, B, 0, stream>>>(ei, e2, m2, s2);
    k_edge2_agg<<<(E_TOT * OUT_DIM + B - 1) / B, B, 0, stream>>>(ei, h2l, e2, s2, out);
}